// GraphOTKuramoto_69965017252358
// MI455X (gfx1250) — compile-verified
//
#include <hip/hip_runtime.h>

// Problem constants (match reference setup_inputs)
#define BATCH 8
#define N 256
#define NF 2048
#define BLK 8
#define EPSV 1e-8f
#define SINK_ITERS 10

typedef __attribute__((ext_vector_type(2))) float v2f;
typedef __attribute__((ext_vector_type(8))) float v8f;

// -----------------------------------------------------------------------------
// Kernel 1: stream A_full [B,2048,2048] once; block-mean-pool 8x8 -> A_lat,
// relu, row-normalize, and K = (W_norm + eps)^10  (== exp(-C/0.1)).
// One block per (b,i) output row; thread j owns one 8x8 block (two b128 loads
// per source row). Writes K[b,i,j] to workspace.
// -----------------------------------------------------------------------------
__global__ void __launch_bounds__(256)
pool_k_kernel(const float* __restrict__ A_full, float* __restrict__ Kmat)
{
    __shared__ float red[N];
    const int bi = blockIdx.x;          // b*N + i
    const int b  = bi >> 8;
    const int i  = bi & (N - 1);
    const int j  = threadIdx.x;

    const long long base = (long long)b * NF * NF + (long long)(i * BLK) * NF + (long long)j * BLK;

    float sum = 0.0f;
#pragma unroll
    for (int r = 0; r < BLK; ++r) {
        const float4* p4 = reinterpret_cast<const float4*>(A_full + base + (long long)r * NF);
        // prefetch the next source row for this thread (speculative; OOB dropped)
        __builtin_prefetch(A_full + base + (long long)(r + 1) * NF, 0, 0);
        float4 x0 = p4[0];
        float4 x1 = p4[1];
        sum += (x0.x + x0.y + x0.z + x0.w) + (x1.x + x1.y + x1.z + x1.w);
    }
    const float mean = sum * (1.0f / 64.0f);
    const float w    = fmaxf(mean, 0.0f);

    // block-wide row sum of relu'd means
    red[j] = w;
    __syncthreads();
    for (int s = 128; s > 0; s >>= 1) {
        if (j < s) red[j] += red[j + s];
        __syncthreads();
    }
    const float rowsum = red[0];

    const float wn = (rowsum > EPSV) ? (w / (rowsum + EPSV)) : (1.0f / (float)N);
    // K = (wn + eps)^10 via exact multiplies
    const float t1 = wn + EPSV;
    const float t2 = t1 * t1;
    const float t4 = t2 * t2;
    const float t8 = t4 * t4;
    Kmat[(long long)bi * N + j] = t8 * t2;
}

// -----------------------------------------------------------------------------
// Kernel 2: per-batch Sinkhorn (10 iters) with WMMA f32 16x16x4 matvecs,
// then pi-normalization + Kuramoto coupling + Euler step.
// One block (256 threads = 8 waves) per batch. u/v/p/theta live in LDS;
// Kmat streams from L2-resident workspace.
// -----------------------------------------------------------------------------
__global__ void __launch_bounds__(256)
sinkhorn_kuramoto_kernel(const float* __restrict__ theta_prev,
                         const float* __restrict__ gamma,
                         const float* __restrict__ omega,
                         const float* __restrict__ alpha,
                         const float* __restrict__ log_kappa,
                         const float* __restrict__ Kmat,
                         float* __restrict__ out)
{
    __shared__ float uS[N], vS[N], pS[N], tS[N], thS[N], red[N];

    const int b    = blockIdx.x;
    const int tid  = threadIdx.x;
    const int wave = tid >> 5;
    const int lane = tid & 31;
    const int hf   = lane >> 4;     // 0 for lanes 0-15, 1 for lanes 16-31
    const int l16  = lane & 15;

    const float* __restrict__ Kb = Kmat + (long long)b * N * N;

    // ---- p = softmax(gamma[b,:]) ----
    const float g = gamma[b * N + tid];
    red[tid] = g;
    __syncthreads();
    for (int s = 128; s > 0; s >>= 1) {
        if (tid < s) red[tid] = fmaxf(red[tid], red[tid + s]);
        __syncthreads();
    }
    const float m = red[0];
    __syncthreads();
    const float e = __expf(g - m);
    red[tid] = e;
    __syncthreads();
    for (int s = 128; s > 0; s >>= 1) {
        if (tid < s) red[tid] += red[tid + s];
        __syncthreads();
    }
    pS[tid]  = e / red[0];
    thS[tid] = theta_prev[b * N + tid];
    uS[tid]  = 1.0f / (float)N;
    vS[tid]  = 1.0f / (float)N;
    __syncthreads();

    // ---- Sinkhorn iterations (matvecs on the WMMA pipe) ----
    for (int it = 0; it < SINK_ITERS; ++it) {
        // t = K v : wave owns row-tiles {wave, wave+8} of 16 rows each.
        for (int rt = wave; rt < 16; rt += 8) {
            const int i0 = rt * 16;
            v8f acc = {};
            for (int kk = 0; kk < N; kk += 4) {
                const int ak = kk + 2 * hf;
                // A tile 16x4: lane half picks K-pair {0,1} or {2,3}
                const float* arow = Kb + (long long)(i0 + l16) * N + ak;
                v2f a;  a.x = arow[0];  a.y = arow[1];
                // B 4x16: v replicated across all 16 columns
                v2f bv; bv.x = vS[ak];  bv.y = vS[ak + 1];
                acc = __builtin_amdgcn_wmma_f32_16x16x4_f32(
                    false, a, false, bv, (short)0, acc, false, false);
            }
            // D column N=0: lane0 holds rows i0..i0+7 (VGPR r), lane16 rows i0+8..i0+15
            if (lane == 0) {
#pragma unroll
                for (int r = 0; r < 8; ++r) tS[i0 + r] = acc[r];
            } else if (lane == 16) {
#pragma unroll
                for (int r = 0; r < 8; ++r) tS[i0 + 8 + r] = acc[r];
            }
        }
        __syncthreads();
        uS[tid] = pS[tid] / (tS[tid] + EPSV);
        __syncthreads();

        // t = K^T u  computed as (u^T K): A rows all = u chunk, B = K tile.
        for (int ct = wave; ct < 16; ct += 8) {
            const int n0 = ct * 16;
            v8f acc = {};
            for (int kk = 0; kk < N; kk += 4) {
                const int ak = kk + 2 * hf;
                v2f a;  a.x = uS[ak];  a.y = uS[ak + 1];   // replicated over rows
                v2f bv; bv.x = Kb[(long long)ak * N + n0 + l16];
                        bv.y = Kb[(long long)(ak + 1) * N + n0 + l16];
                acc = __builtin_amdgcn_wmma_f32_16x16x4_f32(
                    false, a, false, bv, (short)0, acc, false, false);
            }
            // D row M=0 lives in VGPR0 lanes 0-15 (N = lane)
            if (lane < 16) tS[n0 + l16] = acc[0];
        }
        __syncthreads();
        vS[tid] = (1.0f / (float)N) / (tS[tid] + EPSV);
        __syncthreads();
    }

    // ---- pi normalization constant: S = sum_ij u_i K_ij v_j ----
    float si = 0.0f;
    {
        const float* krow = Kb + (long long)tid * N;
        for (int jj = 0; jj < N; ++jj) si += krow[jj] * vS[jj];
        si *= uS[tid];
    }
    red[tid] = si;
    __syncthreads();
    for (int s = 128; s > 0; s >>= 1) {
        if (tid < s) red[tid] += red[tid + s];
        __syncthreads();
    }
    const float inv = 1.0f / (red[0] + EPSV);

    // ---- Kuramoto coupling + Euler step ----
    const float thi = thS[tid];
    const float ui  = uS[tid];
    float cpl = 0.0f;
    {
        const float* krow = Kb + (long long)tid * N;
        const float* arow = alpha + (long long)tid * N;
        for (int jj = 0; jj < N; ++jj) {
            const float pij = ui * krow[jj] * vS[jj] * inv;
            cpl += pij * __sinf(thS[jj] - thi - arow[jj]);
        }
    }
    const float kap = __logf(1.0f + __expf(log_kappa[tid]));   // softplus
    out[b * N + tid] = thi + (omega[tid] + cpl + kap * (g - thi));
}

// -----------------------------------------------------------------------------
// Launch
// -----------------------------------------------------------------------------
extern "C" void kernel_launch(void* const* d_in, const int* in_sizes, int n_in,
                              void* d_out, int out_size, void* d_ws, size_t ws_size,
                              hipStream_t stream)
{
    const float* theta_prev = (const float*)d_in[0];  // [8,256]
    const float* gamma      = (const float*)d_in[1];  // [8,256]
    const float* A_full     = (const float*)d_in[2];  // [8,2048,2048]
    const float* omega      = (const float*)d_in[3];  // [256]
    const float* alpha      = (const float*)d_in[4];  // [256,256]
    const float* log_kappa  = (const float*)d_in[5];  // [256]
    float* out = (float*)d_out;                       // [8,256]

    float* Kws = (float*)d_ws;                        // 8*256*256 floats = 2 MB

    pool_k_kernel<<<dim3(BATCH * N), dim3(N), 0, stream>>>(A_full, Kws);
    sinkhorn_kuramoto_kernel<<<dim3(BATCH), dim3(N), 0, stream>>>(
        theta_prev, gamma, omega, alpha, log_kappa, Kws, out);
}